// KernelGAActivation_506806141067
// MI455X (gfx1250) — compile-verified
//
#include <hip/hip_runtime.h>
#include <hip/hip_bf16.h>

// out[b,f,n] = w0*relu(x[b,f,n]) + (w1+w2)*exp(-sq[n]/0.02)
// sq[n] = sum_{bf=0..63, d=0..1} (x[bf,n] - x[bf, neigh[n,d]])^2
//
// One wave32 per tile of 16 consecutive n. The 128-term reduction per n rides
// the matrix pipe: v_wmma_f32_16x16x4_f32 with B == ones makes each WMMA a
// "row-sum of A accumulated into C". A's 16x4 layout gives lanes 0-15 terms
// K0,K1 (bf=2i, d=0/1) and lanes 16-31 terms K2,K3 (bf=2i+1).
//
// N is passed as a compile-time constant (specialized for the harness's
// N=50000) so the per-iteration row stride folds into the load/store
// instruction's signed 24-bit immediate offset: one base address per
// offset stream instead of per-load v_add_nc_u64 arithmetic.

typedef __attribute__((ext_vector_type(2))) float v2f;
typedef __attribute__((ext_vector_type(8))) float v8f;

#define WAVES_PER_BLOCK 4
#define HALF_BF 32  // BF = 64 rows -> 32 iterations x 2 rows

template <int N_CONST>
__global__ __launch_bounds__(WAVES_PER_BLOCK * 32)
void KernelGAActivation_gfx1250(const float* __restrict__ x,
                                const int* __restrict__ neigh,
                                const float* __restrict__ w,
                                float* __restrict__ out,
                                int N_rt) {
  const int N = (N_CONST > 0) ? N_CONST : N_rt;
  __shared__ int sidx[WAVES_PER_BLOCK][32];

  const int lane = threadIdx.x & 31;
  const int wave = threadIdx.x >> 5;
  const int tile = blockIdx.x * WAVES_PER_BLOCK + wave;
  const int ntiles = (N + 15) >> 4;
  if (tile >= ntiles) return;            // wave-uniform: EXEC stays all-ones

  const int n0   = tile * 16;
  const int m    = lane & 15;            // point-in-tile this lane serves
  const int half = lane >> 4;            // bf parity handled by this lane
  const bool valid = (n0 + m) < N;
  const int n = valid ? (n0 + m) : (N - 1);   // clamp tail loads

  // ---- Stage this wave's 32 neighbor indices into LDS via async copy ----
  {
    unsigned lds_off = (unsigned)(size_t)&sidx[wave][lane];  // low 32b = LDS offset
    long gi = (long)2 * n0 + lane;
    if (gi > (long)2 * N - 1) gi = (long)2 * N - 1;
    const int* gsrc = neigh + gi;
    asm volatile("global_load_async_to_lds_b32 %0, %1, off"
                 :: "v"(lds_off), "v"(gsrc) : "memory");
    asm volatile("s_wait_asynccnt 0" ::: "memory");
  }
  const int j0 = sidx[wave][2 * m + 0];
  const int j1 = sidx[wave][2 * m + 1];

  // Per-lane base addresses (bf parity folded in); all loop loads then use
  // compile-time immediate offsets i * (2*N*4).
  const size_t rowbytes = (size_t)N * 4u;
  const size_t hb = half ? rowbytes : 0;
  const char* __restrict__ xb = (const char*)x;
  const char* __restrict__ pn  = xb + hb + (size_t)(unsigned)n  * 4u;
  const char* __restrict__ pj0 = xb + hb + (size_t)(unsigned)j0 * 4u;
  const char* __restrict__ pj1 = xb + hb + (size_t)(unsigned)j1 * 4u;

  // ---- Accumulate sq over all 64 bf rows via 32 chained WMMAs ----
  v2f bones; bones.x = 1.0f; bones.y = 1.0f;
  v8f c = {};                       // C accumulator starts at 0
  float xs[HALF_BF];                // keep x values for the output pass
  #pragma unroll
  for (int i = 0; i < HALF_BF; ++i) {
    const size_t step = (size_t)i * (2u * rowbytes);   // constexpr when N_CONST>0
    float xv = *(const float*)(pn  + step);  // coalesced per half-wave
    float g0 = *(const float*)(pj0 + step);  // random gather, L2-resident
    float g1 = *(const float*)(pj1 + step);
    xs[i] = xv;
    float d0 = xv - g0;
    float d1 = xv - g1;
    v2f a; a.x = d0 * d0; a.y = d1 * d1;
    // D = A x ones + C  ==> every column of D = row-sum of A (4 terms) + C
    c = __builtin_amdgcn_wmma_f32_16x16x4_f32(
        /*neg_a=*/false, a, /*neg_b=*/false, bones,
        /*c_mod=*/(short)0, c, /*reuse_a=*/false, /*reuse_b=*/false);
  }

  // Lane L now holds sq[(L>=16)*8 + r] in c[r] (columns replicated).
  // Pick r = lane&7, then bpermute so each lane gets sq for its own n.
  const int r = lane & 7;
  float s01 = (r & 1) ? c[1] : c[0];
  float s23 = (r & 1) ? c[3] : c[2];
  float s45 = (r & 1) ? c[5] : c[4];
  float s67 = (r & 1) ? c[7] : c[6];
  float s03 = (r & 2) ? s23 : s01;
  float s47 = (r & 2) ? s67 : s45;
  float sq_held = (r & 4) ? s47 : s03;   // = sq[((lane&16)>>1)|(lane&7)]
  const int src_lane = ((lane & 8) << 1) | (lane & 7);
  float sq_n = __shfl(sq_held, src_lane, 32);

  // kern = exp(-sq / (2*0.1^2)) = exp(-50*sq)
  float kern = __expf(sq_n * -50.0f);

  const float w0 = w[0];
  const float cK = w[1] + w[2];
  const float kterm = cK * kern;

  // ---- Output pass: 64 coalesced non-temporal row stores (imm offsets) ----
  if (valid) {
    char* __restrict__ po = (char*)out + hb + (size_t)(unsigned)n * 4u;
    #pragma unroll
    for (int i = 0; i < HALF_BF; ++i) {
      float xv  = xs[i];
      float rel = xv > 0.0f ? xv : 0.0f;
      float v   = fmaf(w0, rel, kterm);
      __builtin_nontemporal_store(v, (float*)(po + (size_t)i * (2u * rowbytes)));
    }
  }
}

extern "C" void kernel_launch(void* const* d_in, const int* in_sizes, int n_in,
                              void* d_out, int out_size, void* d_ws, size_t ws_size,
                              hipStream_t stream) {
  const float* x     = (const float*)d_in[0];  // (B=2, F=32, N) fp32
  const int*   neigh = (const int*)d_in[1];    // (N, D=2) int32
  const float* w     = (const float*)d_in[2];  // (1, 3) fp32

  float* out = (float*)d_out;                  // (B, F, N) fp32

  const int N = in_sizes[1] / 2;               // neighborhood is N*D
  const int ntiles = (N + 15) / 16;
  const int blocks = (ntiles + WAVES_PER_BLOCK - 1) / WAVES_PER_BLOCK;

  if (N == 50000) {
    hipLaunchKernelGGL((KernelGAActivation_gfx1250<50000>),
                       dim3(blocks), dim3(WAVES_PER_BLOCK * 32), 0, stream,
                       x, neigh, w, out, N);
  } else {
    hipLaunchKernelGGL((KernelGAActivation_gfx1250<0>),
                       dim3(blocks), dim3(WAVES_PER_BLOCK * 32), 0, stream,
                       x, neigh, w, out, N);
  }
}